// VQVectorTokenizerOld_23596550324864
// MI455X (gfx1250) — compile-verified
//
#include <hip/hip_runtime.h>
#include <math.h>

typedef float v2f __attribute__((ext_vector_type(2)));
typedef float v8f __attribute__((ext_vector_type(8)));

constexpr int E  = 256;        // encoder width e
constexpr int DE = 512;        // decoder hidden 2e
constexpr int V  = 2048;       // codebook size
constexpr int TT = 65536;      // total tokens (256 batch * 256 per row)
constexpr long ZE = (long)TT * E;  // z / z_q element count

// ---------------------------------------------------------------- helpers
__device__ __forceinline__ float blockReduce256(float val, float* red, int t) {
    red[t] = val;
    __syncthreads();
    #pragma unroll
    for (int off = 128; off > 0; off >>= 1) {
        if (t < off) red[t] += red[t + off];
        __syncthreads();
    }
    float r = red[0];
    __syncthreads();
    return r;
}

__device__ __forceinline__ float silu_fast(float v) {
    // v * sigmoid(v) with native v_exp_f32
    return v / (1.f + __expf(-v));
}

// ------------------------------------------------ phase 1: code_map(codebook)
// one block per codebook row; produces (c0,c1,c2,||c||^2) bundle per code
__global__ __launch_bounds__(256) void codemap_kernel(
    const float* __restrict__ codebook,
    const float* __restrict__ W1, const float* __restrict__ b1,
    const float* __restrict__ g1, const float* __restrict__ be1,
    const float* __restrict__ W2, const float* __restrict__ b2,
    const float* __restrict__ g2, const float* __restrict__ be2,
    const float* __restrict__ W3, const float* __restrict__ b3,
    float* __restrict__ bundle)
{
    __shared__ float s[E];
    __shared__ float red[E];
    __shared__ float outsh[3];
    const int v = blockIdx.x, t = threadIdx.x;

    float c0 = codebook[v * 3 + 0];
    float c1 = codebook[v * 3 + 1];
    float c2 = codebook[v * 3 + 2];

    // layer 1: d->e, LN, silu
    float h = fmaf(c0, W1[t], fmaf(c1, W1[E + t], fmaf(c2, W1[2 * E + t], b1[t])));
    float sum = blockReduce256(h, red, t);
    float sq  = blockReduce256(h * h, red, t);
    float mu  = sum * (1.f / E);
    float var = sq * (1.f / E) - mu * mu;
    float a1  = silu_fast((h - mu) * rsqrtf(var + 1e-5f) * g1[t] + be1[t]);
    s[t] = a1;
    __syncthreads();

    // layer 2: e->e, LN, silu
    float h2 = b2[t];
    for (int k = 0; k < E; ++k) h2 = fmaf(s[k], W2[k * E + t], h2);
    sum = blockReduce256(h2, red, t);
    sq  = blockReduce256(h2 * h2, red, t);
    mu  = sum * (1.f / E);
    var = sq * (1.f / E) - mu * mu;
    float a2 = silu_fast((h2 - mu) * rsqrtf(var + 1e-5f) * g2[t] + be2[t]);

    // layer 3: e->3 via block reductions
    #pragma unroll
    for (int o = 0; o < 3; ++o) {
        float tot = blockReduce256(a2 * W3[t * 3 + o], red, t);
        if (t == 0) outsh[o] = tot + b3[o];
    }
    __syncthreads();
    if (t == 0) {
        float l0 = outsh[0], l1 = outsh[1], l2 = outsh[2];
        bundle[v * 4 + 0] = l0;
        bundle[v * 4 + 1] = l1;
        bundle[v * 4 + 2] = l2;
        bundle[v * 4 + 3] = l0 * l0 + l1 * l1 + l2 * l2;
    }
}

// ------------------------------------------------ phase 2: z = x @ enc_W + b
// one block per token: x loads broadcast, stores fully coalesced
__global__ __launch_bounds__(256) void encode_x_kernel(
    const float* __restrict__ x, const float* __restrict__ encW,
    const float* __restrict__ encb, float* __restrict__ z)
{
    const int i = blockIdx.x * 256 + threadIdx.x;
    const int T = i >> 8, f = i & 255;
    float x0 = x[T * 3 + 0], x1 = x[T * 3 + 1], x2 = x[T * 3 + 2];
    z[i] = fmaf(x0, encW[f], fmaf(x1, encW[E + f], fmaf(x2, encW[2 * E + f], encb[f])));
}

// ------------------------------------------------ phase 3: nearest code
__global__ __launch_bounds__(256) void argmin_kernel(
    const float* __restrict__ x, const float* __restrict__ bundle,
    int* __restrict__ toks)
{
    __shared__ float4 codes[V];  // 32 KB
    const int t = threadIdx.x;
    const float4* cb = (const float4*)bundle;
    for (int v = t; v < V; v += 256) codes[v] = cb[v];
    __syncthreads();

    const int T = blockIdx.x * 256 + t;
    float x0 = x[T * 3 + 0], x1 = x[T * 3 + 1], x2 = x[T * 3 + 2];
    float best = 3.4e38f;
    int bi = 0;
    for (int v = 0; v < V; ++v) {
        float4 c = codes[v];  // LDS broadcast: all lanes read same address
        float dd = c.w - 2.f * fmaf(x0, c.x, fmaf(x1, c.y, x2 * c.z));
        if (dd < best) { best = dd; bi = v; }
    }
    toks[T] = bi;
}

// ------------------------------------------------ phase 4: z_q = latent[tok] @ enc_W + b
__global__ __launch_bounds__(256) void encode_code_kernel(
    const int* __restrict__ toks, const float* __restrict__ bundle,
    const float* __restrict__ encW, const float* __restrict__ encb,
    float* __restrict__ zq)
{
    const int i = blockIdx.x * 256 + threadIdx.x;
    const int T = i >> 8, f = i & 255;
    const int tok = toks[T];
    float c0 = bundle[tok * 4 + 0], c1 = bundle[tok * 4 + 1], c2 = bundle[tok * 4 + 2];
    zq[i] = fmaf(c0, encW[f], fmaf(c1, encW[E + f], fmaf(c2, encW[2 * E + f], encb[f])));
}

// ------------------------------------------------ phase 5: fused 3-layer decoder
// 32-token tile per block; all intermediates live in 160 KB LDS (320 KB/WGP).
// f32 matrix math on V_WMMA_F32_16X16X4_F32, wave32 fragment layouts per ISA 7.12.2.
// Each wave owns one 16-wide N-tile and BOTH 16-row M-subtiles: every B fragment
// loaded from L2 feeds two WMMAs (halves L2 traffic, doubles WMMA density).
__global__ __launch_bounds__(256) void decoder_kernel(
    const float* __restrict__ zq,
    const float* __restrict__ W1, const float* __restrict__ b1,
    const float* __restrict__ W2, const float* __restrict__ b2,
    const float* __restrict__ W3, const float* __restrict__ b3,
    float* __restrict__ rec)
{
    extern __shared__ float lds[];
    float* act = lds;               // 32 x 256  (32 KB)
    float* gb1 = act + 32 * E;      // 32 x 512  (64 KB)
    float* gb2 = gb1 + 32 * DE;     // 32 x 512  (64 KB)

    const int tid  = threadIdx.x;
    const int wave = tid >> 5;
    const int lane = tid & 31;
    const int lm   = lane & 15;   // M (A) / N (B,D) sub-index
    const int kh   = lane >> 4;   // K-half select
    const int Tb   = blockIdx.x * 32;

    // stage input tile (coalesced)
    for (int i = tid; i < 32 * E; i += 256) act[i] = zq[(long)Tb * E + i];
    __syncthreads();

    // ---- layer 1: [32x256] @ [256x512], silu -> gb1
    for (int nt = wave; nt < 32; nt += 8) {
        const int n0 = nt << 4;
        v8f acc0 = {0.f, 0.f, 0.f, 0.f, 0.f, 0.f, 0.f, 0.f};
        v8f acc1 = {0.f, 0.f, 0.f, 0.f, 0.f, 0.f, 0.f, 0.f};
        const float* ar0 = act + lm * E;          // M-subtile 0
        const float* ar1 = act + (16 + lm) * E;   // M-subtile 1
        for (int kk = 0; kk < E; kk += 4) {
            const int ka = kk + kh * 2;
            v2f b  = { W1[ka * DE + n0 + lm], W1[(ka + 1) * DE + n0 + lm] };
            v2f a0 = { ar0[ka], ar0[ka + 1] };
            v2f a1 = { ar1[ka], ar1[ka + 1] };
            acc0 = __builtin_amdgcn_wmma_f32_16x16x4_f32(false, a0, false, b,
                                                         (short)0, acc0, false, false);
            acc1 = __builtin_amdgcn_wmma_f32_16x16x4_f32(false, a1, false, b,
                                                         (short)0, acc1, false, false);
        }
        const int n = n0 + lm;
        const float bias = b1[n];
        #pragma unroll
        for (int r = 0; r < 8; ++r) {
            gb1[(r + kh * 8) * DE + n]      = silu_fast(acc0[r] + bias);
            gb1[(16 + r + kh * 8) * DE + n] = silu_fast(acc1[r] + bias);
        }
    }
    __syncthreads();

    // ---- layer 2: [32x512] @ [512x512], silu -> gb2
    for (int nt = wave; nt < 32; nt += 8) {
        const int n0 = nt << 4;
        v8f acc0 = {0.f, 0.f, 0.f, 0.f, 0.f, 0.f, 0.f, 0.f};
        v8f acc1 = {0.f, 0.f, 0.f, 0.f, 0.f, 0.f, 0.f, 0.f};
        const float* ar0 = gb1 + lm * DE;
        const float* ar1 = gb1 + (16 + lm) * DE;
        for (int kk = 0; kk < DE; kk += 4) {
            const int ka = kk + kh * 2;
            v2f b  = { W2[ka * DE + n0 + lm], W2[(ka + 1) * DE + n0 + lm] };
            v2f a0 = { ar0[ka], ar0[ka + 1] };
            v2f a1 = { ar1[ka], ar1[ka + 1] };
            acc0 = __builtin_amdgcn_wmma_f32_16x16x4_f32(false, a0, false, b,
                                                         (short)0, acc0, false, false);
            acc1 = __builtin_amdgcn_wmma_f32_16x16x4_f32(false, a1, false, b,
                                                         (short)0, acc1, false, false);
        }
        const int n = n0 + lm;
        const float bias = b2[n];
        #pragma unroll
        for (int r = 0; r < 8; ++r) {
            gb2[(r + kh * 8) * DE + n]      = silu_fast(acc0[r] + bias);
            gb2[(16 + r + kh * 8) * DE + n] = silu_fast(acc1[r] + bias);
        }
    }
    __syncthreads();

    // ---- layer 3: [32x512] @ [512x3] (tiny) -> rec, flat layout rec[T*3+o]
    if (tid < 96) {
        const int tl = tid / 3, o = tid - tl * 3;
        float accv = b3[o];
        const float* grow = gb2 + tl * DE;
        for (int k = 0; k < DE; ++k) accv = fmaf(grow[k], W3[k * 3 + o], accv);
        rec[(long)(Tb + tl) * 3 + o] = accv;
    }
}

// ---------------------------------------------------------------- launcher
extern "C" void kernel_launch(void* const* d_in, const int* in_sizes, int n_in,
                              void* d_out, int out_size, void* d_ws, size_t ws_size,
                              hipStream_t stream) {
    (void)in_sizes; (void)n_in; (void)out_size; (void)ws_size;

    const float* x        = (const float*)d_in[0];
    const float* codebook = (const float*)d_in[1];
    const float* encW     = (const float*)d_in[2];
    const float* encb     = (const float*)d_in[3];
    const float* cmW1     = (const float*)d_in[4];
    const float* cmb1     = (const float*)d_in[5];
    const float* cmg1     = (const float*)d_in[6];
    const float* cmbe1    = (const float*)d_in[7];
    const float* cmW2     = (const float*)d_in[8];
    const float* cmb2     = (const float*)d_in[9];
    const float* cmg2     = (const float*)d_in[10];
    const float* cmbe2    = (const float*)d_in[11];
    const float* cmW3     = (const float*)d_in[12];
    const float* cmb3     = (const float*)d_in[13];
    const float* dW1      = (const float*)d_in[14];
    const float* db1      = (const float*)d_in[15];
    const float* dW2      = (const float*)d_in[16];
    const float* db2      = (const float*)d_in[17];
    const float* dW3      = (const float*)d_in[18];
    const float* db3      = (const float*)d_in[19];

    float* out = (float*)d_out;
    float* z   = out;
    float* zq  = out + ZE;
    float* rec = out + 2 * ZE;

    float* bundle = (float*)d_ws;                                 // 2048 * 4 floats
    int*   toks   = (int*)((char*)d_ws + (size_t)V * 4 * sizeof(float));  // 65536 ints

    codemap_kernel<<<V, 256, 0, stream>>>(codebook, cmW1, cmb1, cmg1, cmbe1,
                                          cmW2, cmb2, cmg2, cmbe2, cmW3, cmb3,
                                          bundle);
    encode_x_kernel<<<(int)(ZE / 256), 256, 0, stream>>>(x, encW, encb, z);
    argmin_kernel<<<TT / 256, 256, 0, stream>>>(x, bundle, toks);
    encode_code_kernel<<<(int)(ZE / 256), 256, 0, stream>>>(toks, bundle, encW, encb, zq);
    decoder_kernel<<<TT / 32, 256, (32 * E + 2 * 32 * DE) * sizeof(float), stream>>>(
        zq, dW1, db1, dW2, db2, dW3, db3, rec);
}